// AnomalyTransformer_31069793419539
// MI455X (gfx1250) — compile-verified
//
#include <hip/hip_runtime.h>
#include <math.h>

// AnomalyTransformer forward for MI455X (gfx1250), fp32 WMMA + TDM path.
// d_in: x[8192,512], Wq,Wk,Wv[512,512], Ws[512,1], W1[512,2048], b1[2048],
//       W2[2048,512], b2[512]
// d_out: x_hat[8192*512] ++ P[8192*8192] ++ S[8192*8192]   (float32)
// d_ws:  Q,K,V [3 x 8192*512] ++ sigma[8192] ++ Z[8192*512] ++ H[8192*2048]
//        = ~134 MB of float scratch.

typedef __attribute__((ext_vector_type(2))) float v2f;
typedef __attribute__((ext_vector_type(8))) float v8f;
typedef unsigned int __attribute__((ext_vector_type(4))) v4u;
typedef int __attribute__((ext_vector_type(4))) v4i;
typedef int __attribute__((ext_vector_type(8))) v8i;

#define NROW   8192
#define DMODEL 512
#define DHID   2048

#if __has_builtin(__builtin_amdgcn_tensor_load_to_lds)
#define USE_TDM 1
#else
#define USE_TDM 0
#endif

#if USE_TDM
// ---------------------------------------------------------------------------
// Tensor Data Mover: 2D tile (tile_d0 x tile_d1 elements, 4B each) from a
// row-major tensor (line stride stride_elems) into LDS at lds_off, with the
// descriptor's LDS padding inserting 2 DWORDs after every 2^(pad_iv+1) DWORDs
// (matches the +2 row pad the compute fragments expect).
// D# packing per CDNA5 ISA (groups 0 and 1;
// groups 2/3 zero -> dims 3/4 unused).
// ---------------------------------------------------------------------------
__device__ __forceinline__ void tdm_load_2d(
    unsigned int lds_off, const void* gptr,
    unsigned int tensor_d0, unsigned int tensor_d1,
    unsigned int tile_d0, unsigned int tile_d1,
    unsigned long long stride_elems,
    unsigned int pad_iv, unsigned int pad_amt)
{
  const unsigned long long ga = (unsigned long long)(uintptr_t)gptr;
  v4u g0;
  g0.x = 1u;                                             // count=1, user mode
  g0.y = lds_off;                                        // lds_addr (bytes)
  g0.z = (unsigned int)ga;                               // global_addr[31:0]
  g0.w = (unsigned int)((ga >> 32) & 0x01FFFFFFull)      // global_addr[56:32]
         | (2u << 30);                                   // type=2 ("image")
  v8i g1;
  g1[0] = (int)((2u << 16)                               // data_size: 4 bytes
                | (1u << 20)                             // pad_enable
                | (pad_iv  << 22)                        // pad_interval code
                | (pad_amt << 25));                      // pad_amount code
  g1[1] = (int)((tensor_d0 & 0xFFFFu) << 16);            // dim0[15:0] (bits 63:48)
  g1[2] = (int)(((tensor_d0 >> 16) & 0xFFFFu)            // dim0[31:16]
                | ((tensor_d1 & 0xFFFFu) << 16));        // dim1[15:0]
  g1[3] = (int)(((tensor_d1 >> 16) & 0xFFFFu)            // dim1[31:16]
                | ((tile_d0 & 0xFFFFu) << 16));          // tile_dim0
  g1[4] = (int)(tile_d1 & 0xFFFFu);                      // tile_dim1 (tile_dim2=0)
  g1[5] = (int)(unsigned int)(stride_elems & 0xFFFFFFFFull);  // dim0_stride[31:0]
  g1[6] = (int)(unsigned int)((stride_elems >> 32) & 0xFFFFull); // stride[47:32]
  g1[7] = 0;                                             // dim1_stride unused (2D)
  v4i z4 = {};
#if defined(__clang_major__) && (__clang_major__ >= 23)
  v8i z8 = {};
  __builtin_amdgcn_tensor_load_to_lds(g0, g1, z4, z4, z8, /*cpol=*/0);
#else
  __builtin_amdgcn_tensor_load_to_lds(g0, g1, z4, z4, /*cpol=*/0);
#endif
}
#endif  // USE_TDM

// ---------------------------------------------------------------------------
// Tiled fp32 WMMA GEMM:  C[M,N] = act(alpha * A[M,K] @ B + bias)
//   TRANSB=0: B is [K,N] row-major.  TRANSB=1: B is [N,K] row-major (B^T);
//             its panel is staged N-major in LDS so fragment reads stay
//             contiguous 8-byte pairs.
// Workgroup: 256 threads = 8 waves. Tile BM=32 x BN=64, K-panel BK=32.
// Wave w owns the 16x16 D tile (wm = w>>2, wn = w&3); 8 x
// v_wmma_f32_16x16x4_f32 per K-panel. Panels staged by the TDM (wave 0 issues,
// s_wait_tensorcnt 0, barrier publishes) or by cooperative loads as fallback.
// M%32==0, N%64==0, K%32==0 holds for every shape here.
// ---------------------------------------------------------------------------
template <bool TRANSB, bool HASBIAS, bool RELU>
__global__ __launch_bounds__(256) void gemm_wmma_f32(
    const float* __restrict__ A, const float* __restrict__ B,
    const float* __restrict__ bias, float* __restrict__ C,
    int M, int Nn, int Kk, float alpha)
{
  constexpr int BM = 32, BN = 64, BK = 32;
  constexpr int LDA_S = BK + 2;                       // 2-DWORD row pad
  constexpr int LDB_S = TRANSB ? (BK + 2) : (BN + 2);
  constexpr int BROWS = TRANSB ? BN : BK;
  __shared__ float As[BM * LDA_S];
  __shared__ float Bs[BROWS * LDB_S];

  const int t    = threadIdx.x;
  const int wave = t >> 5;
  const int lane = t & 31;
  const int wm   = wave >> 2;        // 0..1  (M sub-block)
  const int wn   = wave & 3;         // 0..3  (N sub-block)
  const int tileM = blockIdx.y * BM;
  const int tileN = blockIdx.x * BN;
  const int mn  = lane & 15;         // row (A/D) or col (B/D) within 16
  const int hi  = lane >> 4;         // half-wave selector

  const int lda = Kk;
  const int ldb = TRANSB ? Kk : Nn;

  v8f acc = {};

  for (int k0 = 0; k0 < Kk; k0 += BK) {
#if USE_TDM
    // One DMA per operand panel, issued by wave 0; descriptors are uniform
    // (pure SALU). TENSORcnt is per-wave: wave 0 drains it, then the
    // workgroup barrier publishes the LDS tile to the other waves.
    if (wave == 0) {
      tdm_load_2d((unsigned int)(uintptr_t)As,
                  A + (size_t)tileM * lda + k0,
                  (unsigned int)Kk, (unsigned int)M, BK, BM,
                  (unsigned long long)lda, /*pad every 32 DW*/4u, /*+2 DW*/1u);
      if (TRANSB)
        tdm_load_2d((unsigned int)(uintptr_t)Bs,
                    B + (size_t)tileN * ldb + k0,
                    (unsigned int)Kk, (unsigned int)Nn, BK, BN,
                    (unsigned long long)ldb, 4u, 1u);
      else
        tdm_load_2d((unsigned int)(uintptr_t)Bs,
                    B + (size_t)k0 * ldb + tileN,
                    (unsigned int)Nn, (unsigned int)Kk, BN, BK,
                    (unsigned long long)ldb, /*pad every 64 DW*/5u, 1u);
      __builtin_amdgcn_s_wait_tensorcnt(0);
    }
#else
    // ---- fallback: cooperative staging ----
    {
      const int col = t & 31, row = t >> 5;
#pragma unroll
      for (int i = 0; i < 4; ++i)
        As[(row + 8 * i) * LDA_S + col] =
            A[(size_t)(tileM + row + 8 * i) * lda + (k0 + col)];
    }
    if (TRANSB) {
      const int kc = t & 31, nr = t >> 5;
#pragma unroll
      for (int i = 0; i < 8; ++i)
        Bs[(nr + 8 * i) * LDB_S + kc] =
            B[(size_t)(tileN + nr + 8 * i) * ldb + (k0 + kc)];
    } else {
      const int col = t & 63, row = t >> 6;
#pragma unroll
      for (int i = 0; i < 8; ++i)
        Bs[(row + 4 * i) * LDB_S + col] =
            B[(size_t)(k0 + row + 4 * i) * ldb + (tileN + col)];
    }
    if (k0 + BK < Kk)
      __builtin_prefetch(&A[(size_t)(tileM + (t >> 5)) * lda + (k0 + BK + (t & 31))], 0, 1);
#endif
    __syncthreads();

    // ---- 8 x v_wmma_f32_16x16x4_f32 over the K-panel ----
#pragma unroll
    for (int kk = 0; kk < BK; kk += 4) {
      const int kb = kk + hi * 2;    // ISA 7.12.2: lanes 0-15 K={0,1}, 16-31 K={2,3}
      v2f a, b;
      a.x = As[(wm * 16 + mn) * LDA_S + kb];
      a.y = As[(wm * 16 + mn) * LDA_S + kb + 1];
      if (TRANSB) {
        b.x = Bs[(wn * 16 + mn) * LDB_S + kb];       // B^T staged N-major
        b.y = Bs[(wn * 16 + mn) * LDB_S + kb + 1];
      } else {
        b.x = Bs[kb * LDB_S + (wn * 16 + mn)];
        b.y = Bs[(kb + 1) * LDB_S + (wn * 16 + mn)];
      }
      acc = __builtin_amdgcn_wmma_f32_16x16x4_f32(
          /*neg_a=*/false, a, /*neg_b=*/false, b,
          /*c_mod=*/(short)0, acc, /*reuse_a=*/false, /*reuse_b=*/false);
    }
    __syncthreads();   // all waves done reading before the next panel DMA
  }

  // ---- epilogue: D VGPR r -> row (hi*8 + r), col = lane&15 ----
  const int coln = tileN + wn * 16 + mn;
  const int row0 = tileM + wm * 16 + hi * 8;
  const float bb = HASBIAS ? bias[coln] : 0.0f;
#pragma unroll
  for (int r = 0; r < 8; ++r) {
    float v = alpha * acc[r] + bb;
    if (RELU) v = fmaxf(v, 0.0f);
    C[(size_t)(row0 + r) * Nn + coln] = v;
  }
}

// ---------------------------------------------------------------------------
// sigma[i] = max(x[i,:] . Ws, 0.001). One wave per row.
// ---------------------------------------------------------------------------
__global__ __launch_bounds__(256) void sigma_kernel(
    const float* __restrict__ x, const float* __restrict__ Ws,
    float* __restrict__ sigma)
{
  const int wave = threadIdx.x >> 5, lane = threadIdx.x & 31;
  const int row = blockIdx.x * 8 + wave;
  float s = 0.0f;
  for (int c = lane; c < DMODEL; c += 32)
    s += x[(size_t)row * DMODEL + c] * Ws[c];
#pragma unroll
  for (int off = 16; off > 0; off >>= 1)
    s += __shfl_down(s, off, 32);
  if (lane == 0) sigma[row] = fmaxf(s, 0.001f);
}

// ---------------------------------------------------------------------------
// Prior association. Row-normalization cancels the 1/sqrt(2*pi*sigma) factor:
//   P[i,j] = exp(-0.5*((j-i)/sig_i)^2) / sum_j exp(-0.5*((j-i)/sig_i)^2)
// One block per row; exp recomputed on second pass (cheaper than LDS spill).
// ---------------------------------------------------------------------------
__global__ __launch_bounds__(256) void prior_kernel(
    const float* __restrict__ sigma, float* __restrict__ P)
{
  __shared__ float red[256];
  const int i = blockIdx.x, t = threadIdx.x;
  const float inv = 1.0f / sigma[i];
  float ls = 0.0f;
  for (int j = t; j < NROW; j += 256) {
    const float d = (float)(j - i) * inv;
    ls += __expf(-0.5f * d * d);
  }
  red[t] = ls;
  __syncthreads();
  for (int s = 128; s > 0; s >>= 1) {
    if (t < s) red[t] += red[t + s];
    __syncthreads();
  }
  const float dn = 1.0f / red[0];
  for (int j = t; j < NROW; j += 256) {
    const float d = (float)(j - i) * inv;
    P[(size_t)i * NROW + j] = __expf(-0.5f * d * d) * dn;
  }
}

// ---------------------------------------------------------------------------
// Fused row standardize (unbiased std, eps on std) + softmax, in place.
// Whole 32KB row lives in LDS (WGP has 320KB): one read + one write of S.
// max of standardized row == affine(raw max), so one stats pass suffices.
// ---------------------------------------------------------------------------
__global__ __launch_bounds__(256) void softmax_row_kernel(float* __restrict__ S)
{
  __shared__ float buf[NROW];   // 32 KB
  __shared__ float red[256];
  const int row = blockIdx.x, t = threadIdx.x;
  float* rp = S + (size_t)row * NROW;

  float ls = 0.0f, lq = 0.0f, lm = -INFINITY;
  for (int j = t; j < NROW; j += 256) {
    const float v = rp[j];
    buf[j] = v;
    ls += v; lq += v * v; lm = fmaxf(lm, v);
  }
  red[t] = ls; __syncthreads();
  for (int s = 128; s > 0; s >>= 1) { if (t < s) red[t] += red[t + s]; __syncthreads(); }
  const float sum = red[0]; __syncthreads();
  red[t] = lq; __syncthreads();
  for (int s = 128; s > 0; s >>= 1) { if (t < s) red[t] += red[t + s]; __syncthreads(); }
  const float sq = red[0]; __syncthreads();
  red[t] = lm; __syncthreads();
  for (int s = 128; s > 0; s >>= 1) { if (t < s) red[t] = fmaxf(red[t], red[t + s]); __syncthreads(); }
  const float mx = red[0]; __syncthreads();

  const float mean = sum * (1.0f / NROW);
  const float var  = (sq - sum * mean) * (1.0f / (NROW - 1));   // ddof=1
  const float rstd = 1.0f / (sqrtf(fmaxf(var, 0.0f)) + 1e-5f);
  const float mstd = (mx - mean) * rstd;

  float le = 0.0f;
  for (int j = t; j < NROW; j += 256) {
    const float e = __expf((buf[j] - mean) * rstd - mstd);
    buf[j] = e;
    le += e;
  }
  red[t] = le; __syncthreads();
  for (int s = 128; s > 0; s >>= 1) { if (t < s) red[t] += red[t + s]; __syncthreads(); }
  const float dinv = 1.0f / red[0];
  for (int j = t; j < NROW; j += 256) rp[j] = buf[j] * dinv;
}

// ---------------------------------------------------------------------------
extern "C" void kernel_launch(void* const* d_in, const int* in_sizes, int n_in,
                              void* d_out, int out_size, void* d_ws, size_t ws_size,
                              hipStream_t stream)
{
  (void)in_sizes; (void)n_in; (void)out_size; (void)ws_size;
  const float* x  = (const float*)d_in[0];
  const float* Wq = (const float*)d_in[1];
  const float* Wk = (const float*)d_in[2];
  const float* Wv = (const float*)d_in[3];
  const float* Ws = (const float*)d_in[4];
  const float* W1 = (const float*)d_in[5];
  const float* b1 = (const float*)d_in[6];
  const float* W2 = (const float*)d_in[7];
  const float* b2 = (const float*)d_in[8];

  float* out  = (float*)d_out;
  float* xhat = out;                                   // [8192,512]
  float* P    = xhat + (size_t)NROW * DMODEL;          // [8192,8192]
  float* S    = P + (size_t)NROW * NROW;               // [8192,8192]

  float* ws = (float*)d_ws;
  float* Q  = ws;
  float* K  = Q + (size_t)NROW * DMODEL;
  float* V  = K + (size_t)NROW * DMODEL;
  float* sg = V + (size_t)NROW * DMODEL;
  float* Z  = sg + NROW;
  float* H  = Z + (size_t)NROW * DMODEL;

  const dim3 blk(256);
  const dim3 g_proj(DMODEL / 64, NROW / 32);   // N=512 outputs
  const dim3 g_attn(NROW / 64, NROW / 32);     // N=8192 outputs
  const dim3 g_ffn1(DHID / 64, NROW / 32);     // N=2048 outputs
  const float inv_sqrt_d = 1.0f / sqrtf((float)DMODEL);

  // Q, K, V projections
  hipLaunchKernelGGL((gemm_wmma_f32<false, false, false>), g_proj, blk, 0, stream,
                     x, Wq, nullptr, Q, NROW, DMODEL, DMODEL, 1.0f);
  hipLaunchKernelGGL((gemm_wmma_f32<false, false, false>), g_proj, blk, 0, stream,
                     x, Wk, nullptr, K, NROW, DMODEL, DMODEL, 1.0f);
  hipLaunchKernelGGL((gemm_wmma_f32<false, false, false>), g_proj, blk, 0, stream,
                     x, Wv, nullptr, V, NROW, DMODEL, DMODEL, 1.0f);

  // sigma and prior association
  hipLaunchKernelGGL(sigma_kernel, dim3(NROW / 8), blk, 0, stream, x, Ws, sg);
  hipLaunchKernelGGL(prior_kernel, dim3(NROW), blk, 0, stream, sg, P);

  // S_raw = Q K^T / sqrt(d)  (straight into the S output region)
  hipLaunchKernelGGL((gemm_wmma_f32<true, false, false>), g_attn, blk, 0, stream,
                     Q, K, nullptr, S, NROW, NROW, DMODEL, inv_sqrt_d);

  // standardize + softmax in place
  hipLaunchKernelGGL(softmax_row_kernel, dim3(NROW), blk, 0, stream, S);

  // Z = S V
  hipLaunchKernelGGL((gemm_wmma_f32<false, false, false>), g_proj, blk, 0, stream,
                     S, V, nullptr, Z, NROW, DMODEL, NROW, 1.0f);

  // H = relu(Z W1 + b1)
  hipLaunchKernelGGL((gemm_wmma_f32<false, true, true>), g_ffn1, blk, 0, stream,
                     Z, W1, b1, H, NROW, DHID, DMODEL, 1.0f);

  // x_hat = H W2 + b2
  hipLaunchKernelGGL((gemm_wmma_f32<false, true, false>), g_proj, blk, 0, stream,
                     H, W2, b2, xhat, NROW, DMODEL, DHID, 1.0f);
}